// VariationalGATEncoder_23665269801055
// MI455X (gfx1250) — compile-verified
//
#include <hip/hip_runtime.h>
#include <hip/hip_bf16.h>

typedef __attribute__((ext_vector_type(2))) float v2f;
typedef __attribute__((ext_vector_type(8))) float v8f;

#define C_IN  128
#define C_HID 64
#define C_OUT 32

// ---------------------------------------------------------------------------
// degree / normalization
// ---------------------------------------------------------------------------
__global__ void k_zero(float* __restrict__ p, int n) {
    int i = blockIdx.x * blockDim.x + threadIdx.x;
    if (i < n) p[i] = 0.0f;
}

__global__ void k_deg(const long long* __restrict__ col, float* __restrict__ deg, long long E) {
    long long e = (long long)blockIdx.x * blockDim.x + threadIdx.x;
    if (e < E) unsafeAtomicAdd(&deg[(int)col[e]], 1.0f);
}

// deg includes +1 self-loop, so deg >= 1 always -> dis = rsqrt(deg)
__global__ void k_dis(float* __restrict__ d, int n) {
    int i = blockIdx.x * blockDim.x + threadIdx.x;
    if (i < n) d[i] = __frsqrt_rn(d[i] + 1.0f);
}

// broadcast-init accumulation target with bias: p[i] = b[i % C]
__global__ void k_initb(float* __restrict__ p, const float* __restrict__ b, int C, long long total) {
    long long i = (long long)blockIdx.x * blockDim.x + threadIdx.x;
    if (i < total) p[i] = b[(int)(i & (long long)(C - 1))];
}

// ---------------------------------------------------------------------------
// GEMM1: h_pre[N,64] = x[N,128] @ W1[128,64]   (f32 WMMA 16x16x4)
// 128 threads = 4 waves; each wave does one 16x16 N-tile of a 16-row block.
// ---------------------------------------------------------------------------
__global__ void __launch_bounds__(128) k_gemm1(const float* __restrict__ x,
                                               const float* __restrict__ W,
                                               float* __restrict__ hpre, int N) {
    __shared__ float lA[16 * C_IN];       // 8 KB
    __shared__ float lW[C_IN * C_HID];    // 32 KB
    const int tid = threadIdx.x;
    const int m0  = blockIdx.x * 16;

    for (int i = tid; i < C_IN * C_HID / 4; i += 128)
        ((float4*)lW)[i] = ((const float4*)W)[i];
    const float* xa = x + (size_t)m0 * C_IN;
    for (int i = tid; i < 16 * C_IN / 4; i += 128)
        ((float4*)lA)[i] = ((const float4*)xa)[i];
    __syncthreads();

    const int wave = tid >> 5, lane = tid & 31;
    const int half = lane >> 4, idx = lane & 15;
    const int n0   = wave * 16;

    v8f c = {};
    #pragma unroll
    for (int kb = 0; kb < C_IN; kb += 4) {
        const int ka = kb + 2 * half;
        v2f a, b;
        a[0] = lA[idx * C_IN + ka];
        a[1] = lA[idx * C_IN + ka + 1];
        b[0] = lW[ka * C_HID + n0 + idx];
        b[1] = lW[(ka + 1) * C_HID + n0 + idx];
        c = __builtin_amdgcn_wmma_f32_16x16x4_f32(false, a, false, b, (short)0, c, false, false);
    }
    float* outp = hpre + (size_t)m0 * C_HID + n0 + idx;
    #pragma unroll
    for (int r = 0; r < 8; ++r)
        outp[(size_t)(r + 8 * half) * C_HID] = c[r];
}

// ---------------------------------------------------------------------------
// GEMM2/3 fused: mu_pre = relu(h) @ W2, lv_pre = relu(h) @ W3
// waves 0,1 -> W2 tiles; waves 2,3 -> W3 tiles. ReLU folded into LDS fill.
// ---------------------------------------------------------------------------
__global__ void __launch_bounds__(128) k_gemm23(const float* __restrict__ h,
                                                const float* __restrict__ W2,
                                                const float* __restrict__ W3,
                                                float* __restrict__ mu,
                                                float* __restrict__ lv, int N) {
    __shared__ float lH[16 * C_HID];          // 4 KB
    __shared__ float lW[2 * C_HID * C_OUT];   // 16 KB (W2 then W3)
    const int tid = threadIdx.x;
    const int m0  = blockIdx.x * 16;

    for (int i = tid; i < C_HID * C_OUT / 4; i += 128) {
        ((float4*)lW)[i]                    = ((const float4*)W2)[i];
        ((float4*)(lW + C_HID * C_OUT))[i]  = ((const float4*)W3)[i];
    }
    const float* ha = h + (size_t)m0 * C_HID;
    for (int i = tid; i < 16 * C_HID / 4; i += 128) {
        float4 v = ((const float4*)ha)[i];
        v.x = fmaxf(v.x, 0.0f); v.y = fmaxf(v.y, 0.0f);
        v.z = fmaxf(v.z, 0.0f); v.w = fmaxf(v.w, 0.0f);
        ((float4*)lH)[i] = v;
    }
    __syncthreads();

    const int wave = tid >> 5, lane = tid & 31;
    const int half = lane >> 4, idx = lane & 15;
    const float* lWx = lW + (wave >> 1) * (C_HID * C_OUT);
    const int n0 = (wave & 1) * 16;

    v8f c = {};
    #pragma unroll
    for (int kb = 0; kb < C_HID; kb += 4) {
        const int ka = kb + 2 * half;
        v2f a, b;
        a[0] = lH[idx * C_HID + ka];
        a[1] = lH[idx * C_HID + ka + 1];
        b[0] = lWx[ka * C_OUT + n0 + idx];
        b[1] = lWx[(ka + 1) * C_OUT + n0 + idx];
        c = __builtin_amdgcn_wmma_f32_16x16x4_f32(false, a, false, b, (short)0, c, false, false);
    }
    float* dst = ((wave >> 1) ? lv : mu) + (size_t)m0 * C_OUT + n0 + idx;
    #pragma unroll
    for (int r = 0; r < 8; ++r)
        dst[(size_t)(r + 8 * half) * C_OUT] = c[r];
}

// ---------------------------------------------------------------------------
// Edge scatter, layer 1 (C=64). Work item = (edge_or_selfloop, 4-channel group)
// ---------------------------------------------------------------------------
__global__ void k_scatter64(const long long* __restrict__ row, const long long* __restrict__ col,
                            const float* __restrict__ dis, const float* __restrict__ src,
                            float* __restrict__ dst, long long E, int N) {
    long long idx = (long long)blockIdx.x * blockDim.x + threadIdx.x;
    long long total = (E + (long long)N) << 4;   // 16 groups of 4 channels
    if (idx >= total) return;
    long long e = idx >> 4; int g = (int)(idx & 15);
    int r, c;
    if (e < E) { r = (int)row[e]; c = (int)col[e]; }
    else       { r = c = (int)(e - E); }
    const float norm = dis[r] * dis[c];
    const float4 v = *(const float4*)(src + (size_t)r * C_HID + g * 4);
    float* d = dst + (size_t)c * C_HID + g * 4;
    unsafeAtomicAdd(d + 0, v.x * norm);
    unsafeAtomicAdd(d + 1, v.y * norm);
    unsafeAtomicAdd(d + 2, v.z * norm);
    unsafeAtomicAdd(d + 3, v.w * norm);
}

// ---------------------------------------------------------------------------
// Edge scatter, layers 2+3 fused (C=32 each): edge indices read once.
// ---------------------------------------------------------------------------
__global__ void k_scatter32x2(const long long* __restrict__ row, const long long* __restrict__ col,
                              const float* __restrict__ dis,
                              const float* __restrict__ srcA, const float* __restrict__ srcB,
                              float* __restrict__ dstA, float* __restrict__ dstB,
                              long long E, int N) {
    long long idx = (long long)blockIdx.x * blockDim.x + threadIdx.x;
    long long total = (E + (long long)N) << 3;   // 8 groups of 4 channels
    if (idx >= total) return;
    long long e = idx >> 3; int g = (int)(idx & 7);
    int r, c;
    if (e < E) { r = (int)row[e]; c = (int)col[e]; }
    else       { r = c = (int)(e - E); }
    const float norm = dis[r] * dis[c];
    const float4 va = *(const float4*)(srcA + (size_t)r * C_OUT + g * 4);
    const float4 vb = *(const float4*)(srcB + (size_t)r * C_OUT + g * 4);
    float* da = dstA + (size_t)c * C_OUT + g * 4;
    float* db = dstB + (size_t)c * C_OUT + g * 4;
    unsafeAtomicAdd(da + 0, va.x * norm); unsafeAtomicAdd(da + 1, va.y * norm);
    unsafeAtomicAdd(da + 2, va.z * norm); unsafeAtomicAdd(da + 3, va.w * norm);
    unsafeAtomicAdd(db + 0, vb.x * norm); unsafeAtomicAdd(db + 1, vb.y * norm);
    unsafeAtomicAdd(db + 2, vb.z * norm); unsafeAtomicAdd(db + 3, vb.w * norm);
}

// ---------------------------------------------------------------------------
extern "C" void kernel_launch(void* const* d_in, const int* in_sizes, int n_in,
                              void* d_out, int out_size, void* d_ws, size_t ws_size,
                              hipStream_t stream) {
    const float*     x   = (const float*)d_in[0];
    const long long* ei  = (const long long*)d_in[1];   // int64 edge_index [2,E]
    const float*     W1  = (const float*)d_in[2];
    const float*     b1  = (const float*)d_in[3];
    const float*     W2  = (const float*)d_in[4];
    const float*     b2  = (const float*)d_in[5];
    const float*     W3  = (const float*)d_in[6];
    const float*     b3  = (const float*)d_in[7];
    float* out = (float*)d_out;

    const int       N = in_sizes[0] / C_IN;     // 50000 (divisible by 16)
    const long long E = (long long)in_sizes[1] / 2;
    const long long* row = ei;
    const long long* col = ei + E;

    // workspace layout (floats)
    float* ws = (float*)d_ws;
    size_t o = ((size_t)N + 255) & ~(size_t)255;
    float* dis    = ws;                         // N      (deg, then deg^-1/2)
    float* h_pre  = ws + o;                     // N*64
    float* h      = h_pre  + (size_t)N * C_HID; // N*64
    float* mu_pre = h      + (size_t)N * C_HID; // N*32
    float* lv_pre = mu_pre + (size_t)N * C_OUT; // N*32
    float* mu_out = out;
    float* lv_out = out + (size_t)N * C_OUT;

    const int B = 256;
    // 1. degree + normalization
    k_zero<<<(N + B - 1) / B, B, 0, stream>>>(dis, N);
    k_deg <<<(unsigned)((E + B - 1) / B), B, 0, stream>>>(col, dis, E);
    k_dis <<<(N + B - 1) / B, B, 0, stream>>>(dis, N);

    // 2. layer 1: GEMM -> init bias -> scatter
    k_gemm1<<<N / 16, 128, 0, stream>>>(x, W1, h_pre, N);
    {
        long long t = (long long)N * C_HID;
        k_initb<<<(unsigned)((t + B - 1) / B), B, 0, stream>>>(h, b1, C_HID, t);
        long long work = (E + (long long)N) << 4;
        k_scatter64<<<(unsigned)((work + B - 1) / B), B, 0, stream>>>(row, col, dis, h_pre, h, E, N);
    }

    // 3. layers 2+3: fused GEMMs (ReLU folded) -> init biases -> fused scatter
    k_gemm23<<<N / 16, 128, 0, stream>>>(h, W2, W3, mu_pre, lv_pre, N);
    {
        long long t = (long long)N * C_OUT;
        k_initb<<<(unsigned)((t + B - 1) / B), B, 0, stream>>>(mu_out, b2, C_OUT, t);
        k_initb<<<(unsigned)((t + B - 1) / B), B, 0, stream>>>(lv_out, b3, C_OUT, t);
        long long work = (E + (long long)N) << 3;
        k_scatter32x2<<<(unsigned)((work + B - 1) / B), B, 0, stream>>>(
            row, col, dis, mu_pre, lv_pre, mu_out, lv_out, E, N);
    }
}